// TGEM_73804718015033
// MI455X (gfx1250) — compile-verified
//
#include <hip/hip_runtime.h>
#include <hip/hip_bf16.h>

#define GG   3072
#define BB   2
#define HH   5
#define CCLS 2
#define BH   (BB*HH)
#define LOG2E 1.4426950408889634f

typedef float v2f __attribute__((ext_vector_type(2)));
typedef float v8f __attribute__((ext_vector_type(8)));

// ---------------------------------------------------------------------------
// Kernel 1: q/k/v = x * W (elementwise per head), plus per-(b,h) row-max term
//           m_i = q_i>=0 ? q_i*kmax : q_i*kmin  (exact softmax row max)
// ---------------------------------------------------------------------------
__global__ __launch_bounds__(256) void qkv_kernel(
    const float* __restrict__ x, const float* __restrict__ WQ,
    const float* __restrict__ WK, const float* __restrict__ WV,
    float* __restrict__ q, float* __restrict__ k, float* __restrict__ v,
    float* __restrict__ mx)
{
    const int bh = blockIdx.x;           // 0..9
    const int b = bh / HH, h = bh % HH;
    const int tid = threadIdx.x;
    __shared__ float rmax[256], rmin[256];

    float kmax = -3.4e38f, kmin = 3.4e38f;
    for (int i = tid; i < GG; i += 256) {
        float xv = x[b*GG + i];
        float qi = xv * WQ[h*GG + i];
        float ki = xv * WK[h*GG + i];
        float vi = xv * WV[h*GG + i];
        q[bh*GG + i] = qi; k[bh*GG + i] = ki; v[bh*GG + i] = vi;
        kmax = fmaxf(kmax, ki); kmin = fminf(kmin, ki);
    }
    rmax[tid] = kmax; rmin[tid] = kmin;
    __syncthreads();
    for (int s = 128; s > 0; s >>= 1) {
        if (tid < s) {
            rmax[tid] = fmaxf(rmax[tid], rmax[tid + s]);
            rmin[tid] = fminf(rmin[tid], rmin[tid + s]);
        }
        __syncthreads();
    }
    kmax = rmax[0]; kmin = rmin[0];
    for (int i = tid; i < GG; i += 256) {
        float qi = q[bh*GG + i];
        mx[bh*GG + i] = (qi >= 0.0f) ? qi * kmax : qi * kmin;
    }
}

// ---------------------------------------------------------------------------
// Kernel 2: rank-1 attention row sums via V_WMMA_F32_16X16X4_F32.
//   Each wave owns 16 rows; j-loop steps by 4.
//   A (16x4 f32): lane = half*16 + m, vgpr t -> K = 2*half + t  => e(m, j+K)
//   B (4x16 f32): col0 = v[j+K], col1 = 1.0   => C[:,0]=Sigma e*v, C[:,1]=Sigma e
//   exp(q*k - m) computed as exp2(q2*k - m2) with q2=q*log2e hoisted out of
//   the loop -> inner body is fma + v_exp_f32 + cndmask + wmma only.
//   Epilogue: out_i = (num_i - e_ii*v_i) / den_i  (diag masked post-softmax)
// ---------------------------------------------------------------------------
__global__ __launch_bounds__(256) void attn_kernel(
    const float* __restrict__ q, const float* __restrict__ k,
    const float* __restrict__ v, const float* __restrict__ mx,
    float* __restrict__ outh)
{
    __shared__ __align__(16) float sk[GG];
    __shared__ __align__(16) float sv[GG];
    __shared__ float cbuf[8][32][8];

    const int tid  = threadIdx.x;
    const int bh   = blockIdx.y;
    const int base = bh * GG;

    // stage k, v for this (b,h) into LDS (12 KB + 12 KB)
    for (int i = tid; i < GG; i += 256) { sk[i] = k[base + i]; sv[i] = v[base + i]; }
    __syncthreads();

    const int wave = tid >> 5;            // wave32
    const int lane = tid & 31;
    const int m    = lane & 15;
    const int half = lane >> 4;           // 0 or 1
    const int rowBase = blockIdx.x * 128 + wave * 16;
    const int i_row   = rowBase + m;

    const float q2  = q[base + i_row] * LOG2E;   // fold log2e into invariants
    const float m2  = mx[base + i_row] * LOG2E;
    const bool selv = (m == 0);           // B column 0 carries v
    const bool sel1 = (m == 1);           // B column 1 carries 1.0

    const float2* sk2 = (const float2*)sk;
    const float2* sv2 = (const float2*)sv;

    v8f c = {};
#pragma unroll 4
    for (int j = 0; j < GG; j += 4) {
        const int idx2 = (j >> 1) + half;         // 8B-aligned float2 index
        float2 k2 = sk2[idx2];
        float2 v2 = sv2[idx2];
        v2f a, bb;
        a.x = __builtin_amdgcn_exp2f(fmaf(q2, k2.x, -m2));
        a.y = __builtin_amdgcn_exp2f(fmaf(q2, k2.y, -m2));
        bb.x = selv ? v2.x : (sel1 ? 1.0f : 0.0f);
        bb.y = selv ? v2.y : (sel1 ? 1.0f : 0.0f);
        c = __builtin_amdgcn_wmma_f32_16x16x4_f32(
                /*neg_a=*/false, a, /*neg_b=*/false, bb,
                /*c_mod=*/(short)0, c, /*reuse_a=*/false, /*reuse_b=*/false);
    }

    // unload C through LDS: C[row, col] lives in vgpr row%8, lane (row>=8)*16+col
#pragma unroll
    for (int r = 0; r < 8; ++r) cbuf[wave][lane][r] = c[r];
    __syncthreads();

    if (lane < 16) {                       // lane l handles row m = l
        const int srcLane = (m >= 8) ? 16 : 0;
        const int r = m & 7;
        float num = cbuf[wave][srcLane + 0][r];
        float den = cbuf[wave][srcLane + 1][r];
        float diag = __builtin_amdgcn_exp2f(fmaf(q2, sk[i_row], -m2)) * sv[i_row];
        outh[base + i_row] = (num - diag) / den;
    }
}

// ---------------------------------------------------------------------------
// Kernel 3: combine heads (W0), layernorm (ddof=1, eps added to STD), residual
// ---------------------------------------------------------------------------
__global__ __launch_bounds__(1024) void ln_kernel(
    const float* __restrict__ outh, const float* __restrict__ W0,
    const float* __restrict__ ln_a, const float* __restrict__ ln_b,
    const float* __restrict__ hprev, float* __restrict__ hnext)
{
    const int b = blockIdx.x, tid = threadIdx.x;
    __shared__ float r1[1024], r2[1024];

    float w0[HH];
#pragma unroll
    for (int h = 0; h < HH; ++h) w0[h] = W0[h];

    float o[GG / 1024];
    float s1 = 0.0f, s2 = 0.0f;
#pragma unroll
    for (int cc = 0; cc < GG / 1024; ++cc) {
        int i = tid + cc * 1024;
        float acc = 0.0f;
#pragma unroll
        for (int h = 0; h < HH; ++h)
            acc = fmaf(w0[h], outh[(b*HH + h)*GG + i], acc);
        o[cc] = acc;
        s1 += acc;
        s2 = fmaf(acc, acc, s2);
    }
    r1[tid] = s1; r2[tid] = s2;
    __syncthreads();
    for (int s = 512; s > 0; s >>= 1) {
        if (tid < s) { r1[tid] += r1[tid + s]; r2[tid] += r2[tid + s]; }
        __syncthreads();
    }
    const float S1 = r1[0], S2 = r2[0];
    const float mean = S1 / (float)GG;
    const float var  = (S2 - S1 * S1 / (float)GG) / (float)(GG - 1);
    const float dnm  = sqrtf(fmaxf(var, 0.0f)) + 1e-6f;
#pragma unroll
    for (int cc = 0; cc < GG / 1024; ++cc) {
        int i = tid + cc * 1024;
        hnext[b*GG + i] = hprev[b*GG + i] + ln_a[i] * (o[cc] - mean) / dnm + ln_b[i];
    }
}

// ---------------------------------------------------------------------------
// Kernel 4: final classifier  out[b,c] = h3[b,:] . fc_w[c,:] + fc_b[c]
// ---------------------------------------------------------------------------
__global__ __launch_bounds__(128) void fc_kernel(
    const float* __restrict__ h, const float* __restrict__ fc_w,
    const float* __restrict__ fc_b, float* __restrict__ out)
{
    const int wave = threadIdx.x >> 5, lane = threadIdx.x & 31;
    const int b = wave >> 1, cc = wave & 1;
    float acc = 0.0f;
    for (int g = lane; g < GG; g += 32)
        acc = fmaf(h[b*GG + g], fc_w[cc*GG + g], acc);
#pragma unroll
    for (int off = 16; off > 0; off >>= 1)
        acc += __shfl_down(acc, off, 32);
    if (lane == 0) out[b*CCLS + cc] = acc + fc_b[cc];
}

// ---------------------------------------------------------------------------
extern "C" void kernel_launch(void* const* d_in, const int* in_sizes, int n_in,
                              void* d_out, int out_size, void* d_ws, size_t ws_size,
                              hipStream_t stream)
{
    const float* x    = (const float*)d_in[0];
    const float* WQ[3] = {(const float*)d_in[1], (const float*)d_in[4], (const float*)d_in[7]};
    const float* WK[3] = {(const float*)d_in[2], (const float*)d_in[5], (const float*)d_in[8]};
    const float* WV[3] = {(const float*)d_in[3], (const float*)d_in[6], (const float*)d_in[9]};
    const float* W0[3] = {(const float*)d_in[10], (const float*)d_in[11], (const float*)d_in[12]};
    const float* ln_a = (const float*)d_in[13];
    const float* ln_b = (const float*)d_in[14];
    const float* fc_w = (const float*)d_in[15];
    const float* fc_b = (const float*)d_in[16];

    const size_t BHG = (size_t)BH * GG;
    float* ws  = (float*)d_ws;
    float* q    = ws;
    float* k    = q   + BHG;
    float* v    = k   + BHG;
    float* mx   = v   + BHG;
    float* outh = mx  + BHG;
    float* h1   = outh + BHG;
    float* h2   = h1  + (size_t)BB * GG;

    const float* hcur = x;
    float* hping[2] = {h1, h2};
    for (int layer = 0; layer < 3; ++layer) {
        qkv_kernel<<<BH, 256, 0, stream>>>(hcur, WQ[layer], WK[layer], WV[layer], q, k, v, mx);
        attn_kernel<<<dim3(GG / 128, BH), 256, 0, stream>>>(q, k, v, mx, outh);
        float* hn = hping[layer & 1];
        ln_kernel<<<BB, 1024, 0, stream>>>(outh, W0[layer], ln_a, ln_b, hcur, hn);
        hcur = hn;
    }
    fc_kernel<<<1, 128, 0, stream>>>(hcur, fc_w, fc_b, (float*)d_out);
}